// VectorQuantize_618475291341
// MI455X (gfx1250) — compile-verified
//
#include <hip/hip_runtime.h>
#include <hip/hip_bf16.h>

typedef float v2f __attribute__((ext_vector_type(2)));
typedef float v8f __attribute__((ext_vector_type(8)));

#define NB      16
#define CDIM    256
#define HW      1024           // 32*32
#define NROWS   (NB * HW)      // 16384
#define KCODES  8192
#define ROWS_PER_BLK 32
#define CHUNK   64             // codes per block iteration (2x4 waves of 16x16)
#define QN      (NROWS * CDIM) // quantize elements
#define EPSILON 1e-5f

// ---------- helpers ----------
__device__ __forceinline__ unsigned f_ord(float f) {
    unsigned u = __float_as_uint(f);
    return (u & 0x80000000u) ? ~u : (u | 0x80000000u);   // monotone float->u32
}
__device__ __forceinline__ unsigned long long shfl_xor_u64(unsigned long long v, int m) {
    unsigned lo = (unsigned)v, hi = (unsigned)(v >> 32);
    lo = __shfl_xor(lo, m, 32);
    hi = __shfl_xor(hi, m, 32);
    return ((unsigned long long)hi << 32) | lo;
}

// ---------- kernel 0: zero the commit-loss accumulator ----------
__global__ void vq_zero_ws(float* wsf) {
    if (threadIdx.x == 0 && blockIdx.x == 0) wsf[0] = 0.0f;
}

// ---------- kernel 1: ||e_k||^2  (one wave per codebook row) ----------
__global__ __launch_bounds__(256) void vq_enorm(const float* __restrict__ embed,
                                                float* __restrict__ enorm) {
    int wave = (blockIdx.x * blockDim.x + threadIdx.x) >> 5;
    int lane = threadIdx.x & 31;
    if (wave >= KCODES) return;
    const float* row = embed + (size_t)wave * CDIM;
    float s = 0.0f;
    #pragma unroll
    for (int i = 0; i < CDIM; i += 32) {
        float v = row[i + lane];
        s += v * v;
    }
    #pragma unroll
    for (int m = 16; m >= 1; m >>= 1) s += __shfl_xor(s, m, 32);
    if (lane == 0) enorm[wave] = s;
}

// ---------- kernel 2: main distance GEMM + argmin + gather + commit partial ----------
__global__ __launch_bounds__(256) void vq_main(const float* __restrict__ x,
                                               const float* __restrict__ embed,
                                               const float* __restrict__ enorm,
                                               float* __restrict__ out_q,
                                               float* __restrict__ out_ind,
                                               float* __restrict__ commit_acc) {
    __shared__ float xs[ROWS_PER_BLK][260];            // padded: conflict-free b64 A reads
    __shared__ unsigned long long best[ROWS_PER_BLK];  // (ord(score)<<32)|code
    __shared__ int   codeArr[ROWS_PER_BLK];
    __shared__ float redbuf[8];

    const int tid  = threadIdx.x;
    const int lane = tid & 31;
    const int wave = tid >> 5;
    const int rw   = wave >> 2;        // row-tile  0..1
    const int cw   = wave & 3;         // code-tile 0..3

    const int n0  = blockIdx.x * ROWS_PER_BLK;   // first row of this block
    const int b   = n0 / HW;                     // whole block is inside one image
    const int hw0 = n0 % HW;

    // stage x tile: xs[r][d] = x[b][d][hw0+r]   (coalesced 32-float segments)
    {
        const size_t xbase = (size_t)b * CDIM * HW + hw0;
        for (int idx = tid; idx < ROWS_PER_BLK * CDIM; idx += 256) {
            int d = idx >> 5;
            int r = idx & 31;
            xs[r][d] = x[xbase + (size_t)d * HW + r];
        }
    }
    if (tid < ROWS_PER_BLK) best[tid] = 0xFFFFFFFFFFFFFFFFull;
    __syncthreads();

    const int rowBase = rw * 16;
    const int nsub    = lane & 15;          // column-in-tile / row-in-tile selector
    const int half    = lane >> 4;          // 0: k,k+1   1: k+2,k+3
    const int aRow    = rowBase + nsub;

    for (int cb = 0; cb < KCODES; cb += CHUNK) {
        const int code = cb + cw * 16 + nsub;                 // this lane's B column
        const float* ep = embed + (size_t)code * CDIM;
        const float  en = enorm[code];

        // two independent accumulator chains to break the WMMA RAW-on-C chain
        v8f acc0 = {0.f, 0.f, 0.f, 0.f, 0.f, 0.f, 0.f, 0.f};
        v8f acc1 = {0.f, 0.f, 0.f, 0.f, 0.f, 0.f, 0.f, 0.f};
        #pragma unroll
        for (int k0 = 0; k0 < CDIM; k0 += 8) {
            const int ka = k0 + half * 2;
            const int kb = k0 + 4 + half * 2;
            v2f a0 = *(const v2f*)&xs[aRow][ka];              // ds_load_b64 (hoisted to regs)
            v2f b0 = *(const v2f*)(ep + ka);                  // global_load_b64 (L2-resident)
            acc0 = __builtin_amdgcn_wmma_f32_16x16x4_f32(
                false, a0, false, b0, (short)0, acc0, false, false);
            v2f a1 = *(const v2f*)&xs[aRow][kb];
            v2f b1 = *(const v2f*)(ep + kb);
            acc1 = __builtin_amdgcn_wmma_f32_16x16x4_f32(
                false, a1, false, b1, (short)0, acc1, false, false);
        }

        // score = ||e||^2 - 2*dot  (||x||^2 constant per row, irrelevant for argmin)
        #pragma unroll
        for (int v = 0; v < 8; ++v) {
            int row = rowBase + v + half * 8;
            float s = acc0[v] + acc1[v];
            float score = fmaf(-2.0f, s, en);
            unsigned long long key =
                ((unsigned long long)f_ord(score) << 32) | (unsigned)code;
            // min across the 16 lanes holding this row's 16 candidate codes
            #pragma unroll
            for (int m = 1; m < 16; m <<= 1) {
                unsigned long long o = shfl_xor_u64(key, m);
                if (o < key) key = o;
            }
            if (nsub == 0) atomicMin(&best[row], key);        // ds_min_u64
        }
    }
    __syncthreads();

    // indices
    if (tid < ROWS_PER_BLK) {
        int code = (int)(best[tid] & 0xFFFFFFFFu);
        codeArr[tid] = code;
        out_ind[n0 + tid] = (float)code;
    }
    __syncthreads();

    // gather quantize rows (coalesced) + commit-loss partial
    float local = 0.0f;
    for (int idx = tid; idx < ROWS_PER_BLK * CDIM; idx += 256) {
        int r = idx >> 8;
        int d = idx & 255;
        float e  = embed[(size_t)codeArr[r] * CDIM + d];
        float xv = xs[r][d];
        out_q[(size_t)(n0 + r) * CDIM + d] = e;
        float diff = e - xv;
        local += diff * diff;
    }
    #pragma unroll
    for (int m = 16; m >= 1; m >>= 1) local += __shfl_xor(local, m, 32);
    if (lane == 0) redbuf[wave] = local;
    __syncthreads();
    if (tid == 0) {
        float s = 0.0f;
        #pragma unroll
        for (int w = 0; w < 8; ++w) s += redbuf[w];
        atomicAdd(commit_acc, s);
    }
}

// ---------- kernel 3: perplexity + commit mean ----------
__global__ __launch_bounds__(256) void vq_final(const float* __restrict__ cluster_size,
                                                const float* __restrict__ commit_acc,
                                                float* __restrict__ out) {
    __shared__ float sbuf[8];
    const int tid = threadIdx.x, lane = tid & 31, wave = tid >> 5;
    float s = 0.0f;
    for (int i = tid; i < KCODES; i += 256) {
        float p = cluster_size[i];
        s += p * __logf(p + EPSILON);
    }
    #pragma unroll
    for (int m = 16; m >= 1; m >>= 1) s += __shfl_xor(s, m, 32);
    if (lane == 0) sbuf[wave] = s;
    __syncthreads();
    if (tid == 0) {
        float t = 0.0f;
        #pragma unroll
        for (int w = 0; w < 8; ++w) t += sbuf[w];
        out[QN + NROWS]     = __expf(-t);                        // perplexity
        out[QN + NROWS + 1] = commit_acc[0] / (float)QN;         // commit loss
    }
}

extern "C" void kernel_launch(void* const* d_in, const int* in_sizes, int n_in,
                              void* d_out, int out_size, void* d_ws, size_t ws_size,
                              hipStream_t stream) {
    const float* x            = (const float*)d_in[0];   // (16,256,32,32)
    const float* embed        = (const float*)d_in[1];   // (8192,256)
    const float* cluster_size = (const float*)d_in[2];   // (8192,)
    float* out = (float*)d_out;   // [quantize(QN) | ind(NROWS) | perp | commit]

    float* wsf        = (float*)d_ws;
    float* commit_acc = wsf;            // 1 float
    float* enorm      = wsf + 64;       // 8192 floats (64B-aligned region)

    vq_zero_ws<<<1, 64, 0, stream>>>(wsf);
    vq_enorm<<<KCODES / 8, 256, 0, stream>>>(embed, enorm);
    vq_main<<<NROWS / ROWS_PER_BLK, 256, 0, stream>>>(
        x, embed, enorm, out, out + QN, commit_acc);
    vq_final<<<1, 256, 0, stream>>>(cluster_size, commit_acc, out);
}